// BertBaseLexer_661424964237
// MI455X (gfx1250) — compile-verified
//
#include <hip/hip_runtime.h>
#include <hip/hip_bf16.h>
#include <math.h>

// Problem constants (from reference): B=16, W=256, S=512, F=768, L=4, E=256
#define Bn 16
#define Wn 256
#define Sn 512
#define Fn 768
#define Ln 4
#define En 256
#define NW (Wn - 1)

typedef __attribute__((ext_vector_type(2))) float v2f;
typedef __attribute__((ext_vector_type(4))) float v4f;
typedef __attribute__((ext_vector_type(8))) float v8f;

// One wave (32 lanes) per (b, word-slot) task. 4096 tasks total.
// Per task:
//   out[b,w,0:256]    = emb_table[word_indices[b,w]]           (row gather)
//   out[b,w,256:1024] = mean over span of gamma*softmax(w)·layers  via
//                       V_WMMA_F32_16X16X4_F32 with A = broadcast weight rows,
//                       accumulating span positions through the WMMA C operand.
// start/end are forced into SGPRs (readfirstlane) so the span loop is a
// uniform scalar loop -> EXEC is provably all-ones at every WMMA (ISA req).
__global__ __launch_bounds__(128)
void bert_lexer_fused_kernel(const int* __restrict__ word_indices,
                             const int* __restrict__ span_starts,
                             const int* __restrict__ span_ends,
                             const float* __restrict__ emb_table,
                             const float* __restrict__ layers,
                             const float* __restrict__ layer_weights,
                             const float* __restrict__ gamma,
                             float* __restrict__ out)
{
    const int lane = threadIdx.x & 31;
    const int wave = threadIdx.x >> 5;
    const int task = blockIdx.x * 4 + wave;          // 0 .. B*W-1 (grid sized exactly)
    const int b     = task >> 8;                     // task / W   (W == 256)
    const int wslot = task & (Wn - 1);               // task % W

    float* __restrict__ orow = out + (size_t)task * (En + Fn);

    // ---------------- word-embedding gather: 256 floats, coalesced ----------
    {
        const int widx = word_indices[task];
        const float* __restrict__ erow = emb_table + (size_t)widx * En;
        v4f e0 = *(const v4f*)(erow + lane * 4);
        v4f e1 = *(const v4f*)(erow + 128 + lane * 4);
        *(v4f*)(orow + lane * 4)       = e0;
        *(v4f*)(orow + 128 + lane * 4) = e1;
    }

    float* __restrict__ brow = orow + En;

    if (wslot == 0) {
        // root slot: bert part is zeros (wave-uniform branch)
        v4f z = (v4f){0.f, 0.f, 0.f, 0.f};
        #pragma unroll
        for (int i = 0; i < 6; ++i)
            *(v4f*)(brow + i * 128 + lane * 4) = z;
        return;
    }

    const int wi = wslot - 1;
    // Wave-uniform span bounds: force into SGPRs so the inner loop is a
    // uniform scalar loop (no EXEC-mask loop machinery around the WMMAs).
    const int start = __builtin_amdgcn_readfirstlane(span_starts[b * NW + wi]);
    const int end   = __builtin_amdgcn_readfirstlane(span_ends[b * NW + wi]);
    const int len   = end - start;

    // ---------------- gamma * softmax(layer_weights), computed per-wave -----
    const float lw0 = layer_weights[0], lw1 = layer_weights[1];
    const float lw2 = layer_weights[2], lw3 = layer_weights[3];
    const float mx  = fmaxf(fmaxf(lw0, lw1), fmaxf(lw2, lw3));
    const float x0 = __expf(lw0 - mx), x1 = __expf(lw1 - mx);
    const float x2 = __expf(lw2 - mx), x3 = __expf(lw3 - mx);
    const float scl = gamma[0] / (x0 + x1 + x2 + x3);
    const float w0 = x0 * scl, w1 = x1 * scl, w2 = x2 * scl, w3 = x3 * scl;

    // A matrix (16x4 f32): all 16 rows identical = [w0 w1 w2 w3].
    // ISA layout: VGPR0 = K0 (lanes 0-15) | K2 (lanes 16-31); VGPR1 = K1 | K3.
    v2f a;
    a.x = (lane < 16) ? w0 : w2;
    a.y = (lane < 16) ? w1 : w3;

    // B matrix (4x16 f32) feed: lane supplies layers{0,1} (lanes 0-15) or
    // layers{2,3} (lanes 16-31) at column f = ftile + (lane & 15).
    const size_t BSF = (size_t)Bn * Sn * Fn;
    const int    llo = (lane < 16) ? 0 : 2;
    const int    col = lane & 15;
    // Lane-dependent base for the first span row; step by uniform strides.
    const float* __restrict__ base0 =
        layers + (size_t)llo * BSF + ((size_t)b * Sn + (size_t)start) * Fn + col;

    const float invLen = (len > 0) ? (1.0f / (float)len) : 0.0f;

    #pragma unroll 4
    for (int it = 0; it < Fn / 32; ++it) {           // 24 iterations, 32 f per iter
        const int f0 = it * 32;
        v8f acc0 = {};
        v8f acc1 = {};
        const float* __restrict__ p = base0 + f0;    // layer llo, row `start`
        for (int s = start; s < end; ++s) {          // uniform scalar loop
            v2f bm0, bm1;
            bm0.x = p[0];
            bm0.y = p[BSF];
            bm1.x = p[16];
            bm1.y = p[BSF + 16];
            acc0 = __builtin_amdgcn_wmma_f32_16x16x4_f32(
                false, a, false, bm0, (short)0, acc0, false, false);
            acc1 = __builtin_amdgcn_wmma_f32_16x16x4_f32(
                false, a, false, bm1, (short)0, acc1, false, false);
            p += Fn;                                 // next span row
        }
        // D row M=0 lives in acc0[0] lanes 0-15; acc1[0] lanes 16-31 hold row
        // M=8 == row M=0 of the second f-tile -> blend for one full-wave
        // coalesced 128B store of 32 consecutive f values.
        const float val = ((lane < 16) ? acc0[0] : acc1[0]) * invLen;
        brow[f0 + lane] = val;
    }
}

extern "C" void kernel_launch(void* const* d_in, const int* in_sizes, int n_in,
                              void* d_out, int out_size, void* d_ws, size_t ws_size,
                              hipStream_t stream) {
    const int*   word_indices  = (const int*)d_in[0];
    const int*   span_starts   = (const int*)d_in[1];
    const int*   span_ends     = (const int*)d_in[2];
    const float* emb_table     = (const float*)d_in[3];
    const float* layers        = (const float*)d_in[4];
    const float* layer_weights = (const float*)d_in[5];
    const float* gamma         = (const float*)d_in[6];
    float*       out           = (float*)d_out;

    const int tasks  = Bn * Wn;          // 4096 waves
    const int block  = 128;              // 4 waves per block
    const int grid   = tasks / 4;        // 1024 blocks, exact cover

    bert_lexer_fused_kernel<<<grid, block, 0, stream>>>(
        word_indices, span_starts, span_ends, emb_table, layers,
        layer_weights, gamma, out);
}